// GCN_1219770712387
// MI455X (gfx1250) — compile-verified
//
#include <hip/hip_runtime.h>
#include <hip/hip_bf16.h>

typedef float v2f __attribute__((ext_vector_type(2)));
typedef float v8f __attribute__((ext_vector_type(8)));

#define F_IN  128
#define F_HID 64
#define F_OUT 32

__device__ __forceinline__ void atomAddF(float* p, float v) {
    // lowers to global_atomic_add_f32 (device scope, relaxed)
    __hip_atomic_fetch_add(p, v, __ATOMIC_RELAXED, __HIP_MEMORY_SCOPE_AGENT);
}

// ---------------------------------------------------------------- fill
__global__ void k_fill(float* __restrict__ p, float v, long long n) {
    long long i = (long long)blockIdx.x * blockDim.x + threadIdx.x;
    if (i < n) p[i] = v;
}

// ------------------------------------------------ degree accumulation
__global__ void k_deg(const int* __restrict__ col, float* __restrict__ deg, long long E) {
    long long e = (long long)blockIdx.x * blockDim.x + threadIdx.x;
    if (e < E) atomAddF(&deg[col[e]], 1.0f);
}

// deg -> deg^{-1/2} in place (deg >= 1 always due to self loop)
__global__ void k_dinv(float* __restrict__ deg, int n) {
    int i = blockIdx.x * blockDim.x + threadIdx.x;
    if (i < n) deg[i] = rsqrtf(deg[i]);
}

// ---------------------------------------------------- WMMA fp32 GEMM
// H[n, COUT] = X[n, K] @ W[K, COUT]; one 16x16 output tile per wave.
// A 16x4 frag: lane = M, lanes>=16 hold K+2,K+3.  D: VGPR v = row v / v+8.
template <int K, int COUT>
__global__ void k_gemm_wmma(const float* __restrict__ X, const float* __restrict__ W,
                            float* __restrict__ H, int n) {
    constexpr int COLT = COUT / 16;
    const int lane = threadIdx.x & 31;
    const int wv   = threadIdx.x >> 5;
    const int tile = blockIdx.x * (blockDim.x >> 5) + wv;   // wave-uniform
    const int rowTile = tile / COLT;
    const int colT    = tile - rowTile * COLT;
    const int rowBase = rowTile << 4;
    if (rowBase >= n) return;                               // whole-wave exit

    const int m  = lane & 15;                 // M for A-frag, N for B/D-frag
    const int kh = (lane >> 4) << 1;          // 0 (lanes 0-15) or 2 (lanes 16-31)
    int rowA = rowBase + m;
    if (rowA >= n) rowA = n - 1;              // clamp reads; stores predicated below
    const float*  xr = X + (size_t)rowA * K;
    const float*  wc = W + colT * 16 + m;     // column (colT*16 + m) of W

    v8f acc = {};
    #pragma unroll
    for (int k = 0; k < K; k += 4) {
        v2f a, b;
        float2 av = *(const float2*)(xr + k + kh);   // 8B aligned: k,kh even
        a[0] = av.x;
        a[1] = av.y;
        b[0] = wc[(size_t)(k + kh)     * COUT];
        b[1] = wc[(size_t)(k + kh + 1) * COUT];
        acc = __builtin_amdgcn_wmma_f32_16x16x4_f32(
            /*neg_a=*/false, a, /*neg_b=*/false, b,
            /*c_mod=*/(short)0, acc, /*reuse_a=*/false, /*reuse_b=*/false);
    }

    const int mo = (lane >> 4) << 3;          // +8 rows for upper half-wave
    const int c  = colT * 16 + m;
    #pragma unroll
    for (int v = 0; v < 8; ++v) {
        int r = rowBase + mo + v;
        if (r < n) H[(size_t)r * COUT + c] = acc[v];
    }
}

// -------------------------------------- edge scatter: wave per edge
// out[col] += dinv[row]*dinv[col] * Hs[row]   (F = 64: 2 floats/lane)
__global__ void k_scatter64(const float* __restrict__ Hs, const int* __restrict__ row,
                            const int* __restrict__ col, const float* __restrict__ dinv,
                            float* __restrict__ out, long long E) {
    long long w = ((long long)blockIdx.x * blockDim.x + threadIdx.x) >> 5;
    const int lane = threadIdx.x & 31;
    if (w >= E) return;
    const int r = row[w], c = col[w];
    const float nrm = dinv[r] * dinv[c];
    float2 v = ((const float2*)(Hs + (size_t)r * 64))[lane];   // coalesced 256B/wave
    float* o = out + (size_t)c * 64 + lane * 2;
    atomAddF(o,     v.x * nrm);
    atomAddF(o + 1, v.y * nrm);
}

// (F = 32: 1 float/lane)
__global__ void k_scatter32(const float* __restrict__ Hs, const int* __restrict__ row,
                            const int* __restrict__ col, const float* __restrict__ dinv,
                            float* __restrict__ out, long long E) {
    long long w = ((long long)blockIdx.x * blockDim.x + threadIdx.x) >> 5;
    const int lane = threadIdx.x & 31;
    if (w >= E) return;
    const int r = row[w], c = col[w];
    const float nrm = dinv[r] * dinv[c];
    float v = Hs[(size_t)r * 32 + lane];                        // coalesced 128B/wave
    atomAddF(out + (size_t)c * 32 + lane, v * nrm);
}

// -------------------- fused self-loop + bias (+ optional ReLU) epilogue
// agg[i] = act( agg[i] + dinv[node]^2 * Hs[i] + bias[f] )
template <int F, bool RELU>
__global__ void k_self_bias(float* __restrict__ agg, const float* __restrict__ Hs,
                            const float* __restrict__ dinv, const float* __restrict__ bias,
                            long long total) {
    long long i = (long long)blockIdx.x * blockDim.x + threadIdx.x;
    if (i >= total) return;
    const long long node = i / F;
    const int f = (int)(i - node * F);
    const float di = dinv[node];
    float v = agg[i] + di * di * Hs[i] + bias[f];
    if (RELU) v = fmaxf(v, 0.0f);
    agg[i] = v;
}

// ---------------------------------------------------- link decode
__global__ void k_decode(const float* __restrict__ Z, const int* __restrict__ ea,
                         const int* __restrict__ eb, float* __restrict__ out, long long EL) {
    long long e = (long long)blockIdx.x * blockDim.x + threadIdx.x;
    if (e >= EL) return;
    const float4* za = (const float4*)(Z + (size_t)ea[e] * 32);
    const float4* zb = (const float4*)(Z + (size_t)eb[e] * 32);
    float s = 0.0f;
    #pragma unroll
    for (int i = 0; i < 8; ++i) {
        float4 a = za[i], b = zb[i];
        s += a.x * b.x + a.y * b.y + a.z * b.z + a.w * b.w;
    }
    out[e] = s;
}

static inline int cdiv(long long a, long long b) { return (int)((a + b - 1) / b); }

extern "C" void kernel_launch(void* const* d_in, const int* in_sizes, int n_in,
                              void* d_out, int out_size, void* d_ws, size_t ws_size,
                              hipStream_t stream) {
    const float* x   = (const float*)d_in[0];
    const float* W1  = (const float*)d_in[1];
    const float* b1  = (const float*)d_in[2];
    const float* W2  = (const float*)d_in[3];
    const float* b2  = (const float*)d_in[4];
    const int*   ei  = (const int*)d_in[5];   // [2, E] row-major
    const int*   eli = (const int*)d_in[6];   // [2, EL]

    const long long N  = in_sizes[0] / F_IN;
    const long long E  = in_sizes[5] / 2;
    const long long EL = in_sizes[6] / 2;
    float* out = (float*)d_out;

    // workspace carve-up: (1 + 64 + 64 + 32 + 32) * N floats ≈ 77 MB
    float* deg  = (float*)d_ws;               // becomes dinv in place
    float* h1   = deg  + N;
    float* agg1 = h1   + N * F_HID;
    float* h2   = agg1 + N * F_HID;
    float* agg2 = h2   + N * F_OUT;

    const int* row = ei;       // edge_index[0] (source of message)
    const int* col = ei + E;   // edge_index[1] (destination / segment id)

    // 1) degrees with self loop, then dinv = deg^{-1/2}
    k_fill<<<cdiv(N, 256), 256, 0, stream>>>(deg, 1.0f, N);
    k_deg <<<cdiv(E, 256), 256, 0, stream>>>(col, deg, E);
    k_dinv<<<cdiv(N, 256), 256, 0, stream>>>(deg, (int)N);

    // 2) h1 = x @ W1   (WMMA fp32, 16x16 tile per wave, 8 waves/block)
    {
        long long tiles = ((N + 15) / 16) * (F_HID / 16);
        k_gemm_wmma<F_IN, F_HID><<<cdiv(tiles, 8), 256, 0, stream>>>(x, W1, h1, (int)N);
    }

    // 3) agg1 = scatter(norm * h1[row] -> col); += dinv^2*h1 + b1; ReLU
    k_fill<<<cdiv(N * F_HID, 256), 256, 0, stream>>>(agg1, 0.0f, N * F_HID);
    k_scatter64<<<cdiv(E, 8), 256, 0, stream>>>(h1, row, col, deg, agg1, E);
    k_self_bias<F_HID, true><<<cdiv(N * F_HID, 256), 256, 0, stream>>>(agg1, h1, deg, b1, N * F_HID);

    // 4) h2 = agg1 @ W2
    {
        long long tiles = ((N + 15) / 16) * (F_OUT / 16);
        k_gemm_wmma<F_HID, F_OUT><<<cdiv(tiles, 8), 256, 0, stream>>>(agg1, W2, h2, (int)N);
    }

    // 5) agg2 = scatter(norm * h2[row] -> col); += dinv^2*h2 + b2  (z)
    k_fill<<<cdiv(N * F_OUT, 256), 256, 0, stream>>>(agg2, 0.0f, N * F_OUT);
    k_scatter32<<<cdiv(E, 8), 256, 0, stream>>>(h2, row, col, deg, agg2, E);
    k_self_bias<F_OUT, false><<<cdiv(N * F_OUT, 256), 256, 0, stream>>>(agg2, h2, deg, b2, N * F_OUT);

    // 6) out[e] = dot(z[eli[0][e]], z[eli[1][e]])
    k_decode<<<cdiv(EL, 256), 256, 0, stream>>>(agg2, eli, eli + EL, out, EL);
}